// DistanceLoss_26233660244157
// MI455X (gfx1250) — compile-verified
//
#include <hip/hip_runtime.h>
#include <hip/hip_bf16.h>

typedef float v2f __attribute__((ext_vector_type(2)));
typedef float v8f __attribute__((ext_vector_type(8)));

#define BATCH 2
#define NPTS  8192
#define MPTS  8192
#define MTILES_PER_BATCH (MPTS / 16)
#define WAVES_PER_BLOCK  8

// ---------------------------------------------------------------------------
// Kernel 1: pack reference cloud `input` [B,N,3] into WMMA-B friendly layout:
// packB[b][n] = {x, y, z, -0.5*(x^2+y^2+z^2)}  (float4, 16B aligned)
// ---------------------------------------------------------------------------
__global__ __launch_bounds__(256) void pack_refcloud(
    const float* __restrict__ input, float* __restrict__ packB) {
  int idx = blockIdx.x * blockDim.x + threadIdx.x;   // over B*N
  if (idx >= BATCH * NPTS) return;
  const float* q = input + (size_t)idx * 3;
  float x = q[0], y = q[1], z = q[2];
  float w = -0.5f * (x * x + y * y + z * z);
  float4 v = make_float4(x, y, z, w);
  *(float4*)(packB + (size_t)idx * 4) = v;
}

// ---------------------------------------------------------------------------
// Kernel 2: one wave per 16-row M tile. For each 16-col N tile:
//   C = A x B  with A[m] = [px,py,pz,1], B[:,n] = [qx,qy,qz,-q2/2]
//   => C[m][n] = p.q - q2/2 ;  min_n d2 = p2 - 2*max_n C
// ---------------------------------------------------------------------------
__global__ __launch_bounds__(WAVES_PER_BLOCK * 32) void chamfer_wmma(
    const float* __restrict__ point,   // [B, M, 3]
    const float* __restrict__ packB,   // [B, N, 4]
    float* __restrict__ out)           // [B, M]
{
  const int lane    = threadIdx.x & 31;
  const int wave    = threadIdx.x >> 5;
  const int tile    = blockIdx.x * WAVES_PER_BLOCK + wave;  // exact grid, no tail
  const int b       = tile / MTILES_PER_BATCH;
  const int m0      = (tile % MTILES_PER_BATCH) * 16;
  const int mrow    = lane & 15;
  const int half    = lane >> 4;          // 0: K0/K1, 1: K2/K3

  // ---- A matrix (constant over N scan) ----
  const float* pp = point + ((size_t)b * MPTS + (m0 + mrow)) * 3;
  const float px = pp[0], py = pp[1], pz = pp[2];
  v2f a;
  a.x = half ? pz : px;
  a.y = half ? 1.0f : py;
  const float p2 = px * px + py * py + pz * pz;

  // per-lane base into packed B: column n = lane&15, K-pair select by half
  const float* bbase = packB + ((size_t)b * NPTS + mrow) * 4 + (half << 1);

  v8f cmax;
  #pragma unroll
  for (int i = 0; i < 8; ++i) cmax[i] = -3.402823466e38f;

  // ---- N scan: 512 tiles, unrolled x2 for WMMA pipelining ----
  for (int nt = 0; nt < NPTS / 16; nt += 2) {
    v2f b0 = *(const v2f*)(bbase + (size_t)nt * 64);
    v2f b1 = *(const v2f*)(bbase + (size_t)(nt + 1) * 64);
    v8f c0 = {};
    v8f c1 = {};
    c0 = __builtin_amdgcn_wmma_f32_16x16x4_f32(
        /*neg_a=*/false, a, /*neg_b=*/false, b0,
        /*c_mod=*/(short)0, c0, /*reuse_a=*/false, /*reuse_b=*/false);
    c1 = __builtin_amdgcn_wmma_f32_16x16x4_f32(
        false, a, false, b1, (short)0, c1, false, false);
    #pragma unroll
    for (int i = 0; i < 8; ++i) {
      cmax[i] = fmaxf(cmax[i], fmaxf(c0[i], c1[i]));
    }
  }

  // ---- butterfly max across each 16-lane half (wave32, masks stay in-half) ----
  #pragma unroll
  for (int m = 8; m >= 1; m >>= 1) {
    #pragma unroll
    for (int i = 0; i < 8; ++i) {
      float o = __shfl_xor(cmax[i], m, 32);
      cmax[i] = fmaxf(cmax[i], o);
    }
  }

  // ---- writeback: lanes 0..7 own rows 0..7, lanes 24..31 own rows 8..15 ----
  if (lane < 8) {
    out[(size_t)b * MPTS + m0 + lane] = fmaxf(p2 - 2.0f * cmax[lane], 0.0f);
  } else if (lane >= 24) {
    int r = lane - 24;
    out[(size_t)b * MPTS + m0 + 8 + r] = fmaxf(p2 - 2.0f * cmax[r], 0.0f);
  }
}

extern "C" void kernel_launch(void* const* d_in, const int* in_sizes, int n_in,
                              void* d_out, int out_size, void* d_ws, size_t ws_size,
                              hipStream_t stream) {
  const float* input = (const float*)d_in[0];   // [B, N, 3]
  const float* point = (const float*)d_in[1];   // [B, M, 3]
  float* out = (float*)d_out;                   // [B, M]
  float* packB = (float*)d_ws;                  // [B, N, 4] = 256 KB

  // pack reference cloud
  {
    int total = BATCH * NPTS;
    pack_refcloud<<<(total + 255) / 256, 256, 0, stream>>>(input, packB);
  }
  // main WMMA scan: 1024 M-tiles, 8 waves/block
  {
    int tiles = BATCH * MTILES_PER_BATCH;
    chamfer_wmma<<<tiles / WAVES_PER_BLOCK, WAVES_PER_BLOCK * 32, 0, stream>>>(
        point, packB, out);
  }
}